// DAGLayer_45732811767838
// MI455X (gfx1250) — compile-verified
//
#include <hip/hip_runtime.h>

typedef __attribute__((ext_vector_type(2))) float v2f;
typedef __attribute__((ext_vector_type(8))) float v8f;

#define Nn  32      // batch
#define Cc  256     // channels
#define Ll  256     // leaves
#define Mm  512     // computed nodes
#define Pp  8       // fan-in
#define Kk  2048    // Cc*Pp : GEMM K
#define TOT 768     // Ll+Mm
#define NC  8192    // Nn*Cc
#define NWG 16      // one workgroup per 16-wide out-channel tile

// ---------------------------------------------------------------------------
// Init: copy leaves into d_out [N,C,TOT] and into the transposed gather
// buffer wsT [TOT][N*C]; reset the grid-barrier counter.
// ---------------------------------------------------------------------------
__global__ __launch_bounds__(256) void dag_init(const float* __restrict__ x,
                                                float* __restrict__ out,
                                                float* __restrict__ wsT,
                                                unsigned* __restrict__ bar) {
  if (blockIdx.x == 0 && threadIdx.x == 0) *bar = 0u;
  size_t t = (size_t)blockIdx.x * 256 + threadIdx.x;   // over N*C*L elements
  unsigned l  = (unsigned)(t & (Ll - 1));
  unsigned nc = (unsigned)(t >> 8);                    // n*C + c
  float v = x[t];
  out[(size_t)nc * TOT + l] = v;
  wsT[(size_t)l * NC + nc]  = v;
}

// ---------------------------------------------------------------------------
// Persistent DAG kernel: 16 wgs x 256 threads. wg = one 16-wide o-tile.
// Wave w handles K-slice [w*256, w*256+256) for BOTH 16-row batch tiles.
// Per node: WMMA-accumulate, LDS reduce over 8 waves, bias, write column,
// device-scope release/acquire barrier.
// ---------------------------------------------------------------------------
__global__ __launch_bounds__(256) void dag_nodes(const float* __restrict__ W,
                                                 const float* __restrict__ b,
                                                 const int* __restrict__ parents,
                                                 float* __restrict__ out,
                                                 float* __restrict__ wsT,
                                                 unsigned* __restrict__ bar) {
  __shared__ v8f red[8][2][32];   // [wave][mtile][lane] partial D tiles (16 KB)
  __shared__ int par[Pp];

  const unsigned tid   = threadIdx.x;
  const unsigned wave  = tid >> 5;
  const unsigned lane  = tid & 31;
  const unsigned o0    = blockIdx.x * 16;      // this wg's out-channel tile
  const unsigned m0    = lane & 15;            // A row (lanes 0-15), B col
  const unsigned khalf = (lane >> 4) << 1;     // K sub-offset per lane half
  const unsigned kw0   = wave * 256;           // this wave's K base
  const unsigned ocol  = o0 + m0;              // B-matrix column owned by lane

  for (int i = 0; i < Mm; ++i) {
    if (tid < Pp) par[tid] = parents[i * Pp + tid];
    __syncthreads();

    const float* Wr = W + ((size_t)i * Cc + ocol) * Kk;   // B[k][ocol] = Wr[k]
    if (i + 1 < Mm)  // overlap next node's 1GB weight stream with this node
      __builtin_prefetch(W + ((size_t)(i + 1) * Cc + ocol) * Kk + kw0, 0, 0);

    v8f accA = (v8f)0.0f, accB = (v8f)0.0f;   // mtile 0 (batch rows 0-15)
    v8f accC = (v8f)0.0f, accD = (v8f)0.0f;   // mtile 1 (batch rows 16-31)

#pragma unroll 4
    for (int t = 0; t < 64; ++t) {
      unsigned kk = kw0 + (unsigned)t * 4 + khalf;
      // B fragment: contiguous in W's (c,p) order -> v2f, non-temporal stream
      v2f bv = __builtin_nontemporal_load((const v2f*)(Wr + kk));
      // A fragment: A[m][k] = wsT[parents[k&7]][m*C + (k>>3)]  (L2-resident)
      unsigned k1 = kk, k2 = kk + 1;
      const float* col1 = wsT + (size_t)par[k1 & 7] * NC + (k1 >> 3);
      const float* col2 = wsT + (size_t)par[k2 & 7] * NC + (k2 >> 3);
      v2f a0 = { col1[m0 * Cc],        col2[m0 * Cc]        };
      v2f a1 = { col1[(m0 + 16) * Cc], col2[(m0 + 16) * Cc] };
      if (t & 1) {
        accB = __builtin_amdgcn_wmma_f32_16x16x4_f32(false, a0, false, bv,
                                                     (short)0, accB, false, false);
        accD = __builtin_amdgcn_wmma_f32_16x16x4_f32(false, a1, false, bv,
                                                     (short)0, accD, false, false);
      } else {
        accA = __builtin_amdgcn_wmma_f32_16x16x4_f32(false, a0, false, bv,
                                                     (short)0, accA, false, false);
        accC = __builtin_amdgcn_wmma_f32_16x16x4_f32(false, a1, false, bv,
                                                     (short)0, accC, false, false);
      }
    }
    red[wave][0][lane] = accA + accB;
    red[wave][1][lane] = accC + accD;
    __syncthreads();

    // Waves 0,1 reduce the 8 K-partials and emit the [32 x 16] column tile.
    if (wave < 2) {
      const unsigned mt = wave;
      v8f s = red[0][mt][lane];
#pragma unroll
      for (int w = 1; w < 8; ++w) s += red[w][mt][lane];
      float bb = b[(size_t)i * Cc + ocol];
      unsigned rowhi = (lane >> 4) * 8;
#pragma unroll
      for (int v = 0; v < 8; ++v) {
        unsigned mb = mt * 16 + rowhi + (unsigned)v;   // batch index
        float y = s[v] + bb;
        out[((size_t)mb * Cc + ocol) * TOT + (Ll + i)] = y;
        wsT[(size_t)(Ll + i) * NC + mb * Cc + ocol]    = y;
      }
    }

    // Device-wide release/acquire barrier so node i+1 can gather column L+i.
    __builtin_amdgcn_fence(__ATOMIC_RELEASE, "agent");
    __syncthreads();
    if (tid == 0) {
      atomicAdd(bar, 1u);
      unsigned target = (unsigned)(i + 1) * NWG;
      while (__hip_atomic_load(bar, __ATOMIC_RELAXED, __HIP_MEMORY_SCOPE_AGENT) < target)
        __builtin_amdgcn_s_sleep(1);
    }
    __syncthreads();
    __builtin_amdgcn_fence(__ATOMIC_ACQUIRE, "agent");
  }
}

// ---------------------------------------------------------------------------
extern "C" void kernel_launch(void* const* d_in, const int* in_sizes, int n_in,
                              void* d_out, int out_size, void* d_ws, size_t ws_size,
                              hipStream_t stream) {
  const float* x       = (const float*)d_in[0];   // [N,C,L]
  const float* W       = (const float*)d_in[1];   // [M,C,C,P]
  const float* b       = (const float*)d_in[2];   // [M,C]
  const int*   parents = (const int*)d_in[3];     // [M,P]
  float* out = (float*)d_out;                     // [N,C,L+M]

  float*    wsT = (float*)d_ws;                                        // [TOT][N*C]
  unsigned* bar = (unsigned*)((char*)d_ws +
                              (size_t)TOT * NC * sizeof(float));       // barrier ctr

  dag_init<<<(Nn * Cc * Ll) / 256, 256, 0, stream>>>(x, out, wsT, bar);
  dag_nodes<<<NWG, 256, 0, stream>>>(W, b, parents, out, wsT, bar);
}